// GCN_50543175139618
// MI455X (gfx1250) — compile-verified
//
#include <hip/hip_runtime.h>
#include <hip/hip_bf16.h>
#include <math.h>

// ---------------------------------------------------------------------------
// GCN pipeline for B=64, N=133, feats 75->150->256->128->75->1 on gfx1250.
// Matrix work uses V_WMMA_F32_16X16X4_F32 (fp32 in/out, wave32), with each
// wave register-blocked to a 32x32 output tile (4 accumulators) so the
// WMMA:VMEM ratio is 4:6 per K-step instead of 1:3.
// ---------------------------------------------------------------------------

typedef __attribute__((ext_vector_type(2))) float v2f;
typedef __attribute__((ext_vector_type(8))) float v8f;

#define GB    64      // batch
#define NODES 133     // real nodes
#define NP    160     // padded nodes (10 * 16, 5 * 32)
#define LDH   256     // activation row stride (floats)

// ---------------- pad/copy: src (B,R,C) contiguous -> dst (B,Rp,ldd), cols [0,Cp) zero-padded
__global__ void gcn_pad_copy(const float* __restrict__ src, float* __restrict__ dst,
                             int R, int C, int Rp, int Cp, int ldd) {
  int b = blockIdx.y;
  long long idx = (long long)blockIdx.x * blockDim.x + threadIdx.x;
  if (idx >= (long long)Rp * Cp) return;
  int r = (int)(idx / Cp);
  int c = (int)(idx % Cp);
  float v = (r < R && c < C) ? src[((long long)b * R + r) * C + c] : 0.0f;
  dst[((long long)b * Rp + r) * ldd + c] = v;
}

// ---------------- fp32 WMMA GEMM: C = act(A(M,K) @ B(K,F) + bias)
// One wave (32 threads) per 32x32 output tile (2x2 subtiles of 16x16).
// grid = (M/32, F/32, batches).
// A-matrix 16x4 fp32 layout:  lane<16: K=0,1 ; lane>=16: K=2,3 ; M = lane&15
// B-matrix 4x16 fp32 layout:  lane<16: K=0,1 ; lane>=16: K=2,3 ; N = lane&15
// C/D 16x16 fp32 layout:      vgpr v: M = v + (lane>=16?8:0), N = lane&15
__global__ __launch_bounds__(32)
void gcn_wmma_gemm_f32(const float* __restrict__ A, int lda, long long strideAb,
                       const float* __restrict__ Bm, int ldb, long long strideBb,
                       float* __restrict__ C, int ldc, long long strideCb,
                       const float* __restrict__ bias, int Ksteps, int relu) {
  int mt = blockIdx.x, nt = blockIdx.y, bz = blockIdx.z;
  int lane = threadIdx.x;
  int half = lane >> 4;     // 0 or 1
  int l16  = lane & 15;

  const float* Ab = A  + (long long)bz * strideAb + (long long)mt * 32 * lda;
  const float* Bb = Bm + (long long)bz * strideBb + nt * 32;
  float*       Cb = C  + (long long)bz * strideCb + (long long)mt * 32 * ldc + nt * 32;

  const float* a0p = Ab + (long long)l16 * lda + half * 2;     // rows m0..m0+15
  const float* a1p = a0p + (long long)16 * lda;                // rows m0+16..m0+31
  const float* b0p = Bb + l16 + (long long)(half * 2) * ldb;   // cols n0..n0+15
  const float* b1p = b0p + 16;                                 // cols n0+16..n0+31

  v8f acc00 = {}, acc01 = {}, acc10 = {}, acc11 = {};
  for (int ks = 0; ks < Ksteps; ++ks) {
    v2f a0 = *(const v2f*)a0p;
    v2f a1 = *(const v2f*)a1p;
    v2f b0, b1;
    b0.x = b0p[0];  b0.y = b0p[ldb];
    b1.x = b1p[0];  b1.y = b1p[ldb];
    acc00 = __builtin_amdgcn_wmma_f32_16x16x4_f32(false, a0, false, b0, (short)0, acc00, false, false);
    acc01 = __builtin_amdgcn_wmma_f32_16x16x4_f32(false, a0, false, b1, (short)0, acc01, false, false);
    acc10 = __builtin_amdgcn_wmma_f32_16x16x4_f32(false, a1, false, b0, (short)0, acc10, false, false);
    acc11 = __builtin_amdgcn_wmma_f32_16x16x4_f32(false, a1, false, b1, (short)0, acc11, false, false);
    a0p += 4;  a1p += 4;
    b0p += (long long)4 * ldb;  b1p += (long long)4 * ldb;
  }

  float bv0 = bias ? bias[nt * 32 + l16]      : 0.0f;
  float bv1 = bias ? bias[nt * 32 + 16 + l16] : 0.0f;
#pragma unroll
  for (int v = 0; v < 8; ++v) {
    int row = v + half * 8;
    float c00 = acc00[v] + bv0;
    float c01 = acc01[v] + bv1;
    float c10 = acc10[v] + bv0;
    float c11 = acc11[v] + bv1;
    if (relu) {
      c00 = fmaxf(c00, 0.0f); c01 = fmaxf(c01, 0.0f);
      c10 = fmaxf(c10, 0.0f); c11 = fmaxf(c11, 0.0f);
    }
    Cb[(long long)row * ldc + l16]             = c00;
    Cb[(long long)row * ldc + 16 + l16]        = c01;
    Cb[(long long)(row + 16) * ldc + l16]      = c10;
    Cb[(long long)(row + 16) * ldc + 16 + l16] = c11;
  }
}

// ---------------- neighbor max-pooling: out[b,i,f] = max_{j: adj[b,i,j]>1e-5} h[b,j,f] (0 if none)
__global__ __launch_bounds__(256)
void gcn_pool(const float* __restrict__ h, const float* __restrict__ adjP,
              float* __restrict__ out, int F) {
  int i = blockIdx.x;       // padded node row (0..NP-1)
  int b = blockIdx.y;
  const float* arow = adjP + ((long long)b * NP + i) * NP;

  __shared__ int nbrs[NODES];
  __shared__ int cnt;
  if (threadIdx.x == 0) cnt = 0;
  __syncthreads();
  for (int j = threadIdx.x; j < NODES; j += blockDim.x) {
    if (arow[j] > 1e-5f) {
      int p = atomicAdd(&cnt, 1);
      nbrs[p] = j;
    }
  }
  __syncthreads();
  int n = cnt;

  const float* hb   = h + (long long)b * NP * LDH;
  float*       orow = out + ((long long)b * NP + i) * LDH;
  for (int f = threadIdx.x; f < F; f += blockDim.x) {
    float m = -INFINITY;
    for (int t = 0; t < n; ++t) m = fmaxf(m, hb[(long long)nbrs[t] * LDH + f]);
    orow[f] = (n > 0) ? m : 0.0f;
  }
}

// ---------------- head: gc4 (75->1) + adj mix + fc1 (132->3) + fin (4->1) + sigmoid
__global__ __launch_bounds__(256)
void gcn_head(const float* __restrict__ h3p, const float* __restrict__ adjP,
              const float* __restrict__ w4, const float* __restrict__ b4,
              const float* __restrict__ fc1w, const float* __restrict__ fc1b,
              const float* __restrict__ finw, const float* __restrict__ finb,
              float* __restrict__ out) {
  int b = blockIdx.x;
  __shared__ float z[NODES];
  __shared__ float y[NODES];
  __shared__ float rest[3];
  int t = threadIdx.x;

  for (int j = t; j < NODES; j += blockDim.x) {
    const float* row = h3p + ((long long)b * NP + j) * LDH;
    float s = 0.0f;
    for (int f = 0; f < 75; ++f) s += row[f] * w4[f];
    z[j] = s;
  }
  __syncthreads();
  for (int i = t; i < NODES; i += blockDim.x) {
    const float* ar = adjP + ((long long)b * NP + i) * NP;
    float s = b4[0];
    for (int j = 0; j < NODES; ++j) s += ar[j] * z[j];
    y[i] = fmaxf(s, 0.0f);
  }
  __syncthreads();
  if (t < 3) {
    float s = fc1b[t];
    for (int k = 0; k < NODES - 1; ++k) s += y[1 + k] * fc1w[k * 3 + t];
    rest[t] = s;
  }
  __syncthreads();
  if (t == 0) {
    float s = finb[0] + y[0] * finw[0]
            + rest[0] * finw[1] + rest[1] * finw[2] + rest[2] * finw[3];
    out[b] = 1.0f / (1.0f + expf(-s));
  }
}

// ---------------------------------------------------------------------------
extern "C" void kernel_launch(void* const* d_in, const int* in_sizes, int n_in,
                              void* d_out, int out_size, void* d_ws, size_t ws_size,
                              hipStream_t stream) {
  const float* x     = (const float*)d_in[0];   // (64,133,75)
  const float* adj   = (const float*)d_in[1];   // (64,133,133)
  const float* emb_w = (const float*)d_in[2];   // (75,150)
  const float* emb_b = (const float*)d_in[3];   // (150)
  const float* gc1_w = (const float*)d_in[4];   // (150,256)
  const float* gc1_b = (const float*)d_in[5];   // (256)
  const float* gc2_w = (const float*)d_in[6];   // (256,128)
  const float* gc2_b = (const float*)d_in[7];   // (128)
  const float* gc3_w = (const float*)d_in[8];   // (128,75)
  const float* gc3_b = (const float*)d_in[9];   // (75)
  const float* gc4_w = (const float*)d_in[10];  // (75)
  const float* gc4_b = (const float*)d_in[11];  // (1)
  const float* fc1_w = (const float*)d_in[12];  // (132,3)
  const float* fc1_b = (const float*)d_in[13];  // (3)
  const float* fin_w = (const float*)d_in[14];  // (4)
  const float* fin_b = (const float*)d_in[15];  // (1)
  float* out = (float*)d_out;                    // (64)

  // workspace layout (floats)
  const long long OFF_ADJP = 0;                              // 64*160*160
  const long long OFF_BUFA = OFF_ADJP + (long long)GB*NP*NP; // 64*160*256
  const long long OFF_BUFB = OFF_BUFA + (long long)GB*NP*LDH;
  const long long OFF_EMBW = OFF_BUFB + (long long)GB*NP*LDH; // 76*160
  const long long OFF_GC1W = OFF_EMBW + 76*160;               // 152*256
  const long long OFF_GC3W = OFF_GC1W + 152*256;              // 128*96
  const long long OFF_EMBB = OFF_GC3W + 128*96;               // 160
  const long long OFF_GC3B = OFF_EMBB + 160;                  // 96
  const long long TOTAL    = OFF_GC3B + 96;
  if (ws_size < (size_t)TOTAL * sizeof(float)) return;

  float* ws     = (float*)d_ws;
  float* adjP   = ws + OFF_ADJP;
  float* bufA   = ws + OFF_BUFA;
  float* bufB   = ws + OFF_BUFB;
  float* emb_wp = ws + OFF_EMBW;
  float* gc1_wp = ws + OFF_GC1W;
  float* gc3_wp = ws + OFF_GC3W;
  float* emb_bp = ws + OFF_EMBB;
  float* gc3_bp = ws + OFF_GC3B;

  // --- padding passes
  gcn_pad_copy<<<dim3((NP*NP + 255)/256, GB), 256, 0, stream>>>(adj,  adjP,  133, 133, NP, NP, NP);
  gcn_pad_copy<<<dim3((NP*80 + 255)/256, GB), 256, 0, stream>>>(x,    bufA,  133,  75, NP, 80, LDH);
  gcn_pad_copy<<<dim3((76*160 + 255)/256, 1), 256, 0, stream>>>(emb_w, emb_wp, 75, 150,  76, 160, 160);
  gcn_pad_copy<<<dim3((152*256 + 255)/256, 1), 256, 0, stream>>>(gc1_w, gc1_wp, 150, 256, 152, 256, 256);
  gcn_pad_copy<<<dim3((128*96 + 255)/256, 1), 256, 0, stream>>>(gc3_w, gc3_wp, 128,  75, 128,  96,  96);
  gcn_pad_copy<<<dim3(1, 1), 256, 0, stream>>>(emb_b, emb_bp, 1, 150, 1, 160, 160);
  gcn_pad_copy<<<dim3(1, 1), 256, 0, stream>>>(gc3_b, gc3_bp, 1,  75, 1,  96,  96);

  const int MT = (GB * NP) / 32;  // 320 row tiles (32 rows each) for dense stages
  const int AT = NP / 32;         // 5 row tiles for adj stages
  const long long SA = (long long)NP * NP;   // adjP batch stride
  const long long SH = (long long)NP * LDH;  // activation batch stride

  // S1: bufB = relu(bufA(x) @ emb_wp + emb_b)   M=10240 K=76  F=160
  gcn_wmma_gemm_f32<<<dim3(MT, 5, 1), 32, 0, stream>>>(bufA, LDH, 0, emb_wp, 160, 0,
                                                       bufB, LDH, 0, emb_bp, 19, 1);
  // S2: bufA = bufB @ gc1_wp                    K=152 F=256
  gcn_wmma_gemm_f32<<<dim3(MT, 8, 1), 32, 0, stream>>>(bufB, LDH, 0, gc1_wp, 256, 0,
                                                       bufA, LDH, 0, nullptr, 38, 0);
  // S3: bufB = relu(adjP @ bufA + gc1_b)        batched, K=160 F=256
  gcn_wmma_gemm_f32<<<dim3(AT, 8, GB), 32, 0, stream>>>(adjP, NP, SA, bufA, LDH, SH,
                                                        bufB, LDH, SH, gc1_b, 40, 1);
  // S4: pool -> bufA (F=256)
  gcn_pool<<<dim3(NP, GB), 256, 0, stream>>>(bufB, adjP, bufA, 256);
  // S5: bufB = bufA @ gc2_w                     K=256 F=128 (no padding needed)
  gcn_wmma_gemm_f32<<<dim3(MT, 4, 1), 32, 0, stream>>>(bufA, LDH, 0, gc2_w, 128, 0,
                                                       bufB, LDH, 0, nullptr, 64, 0);
  // S6: bufA = relu(adjP @ bufB + gc2_b)        F=128
  gcn_wmma_gemm_f32<<<dim3(AT, 4, GB), 32, 0, stream>>>(adjP, NP, SA, bufB, LDH, SH,
                                                        bufA, LDH, SH, gc2_b, 40, 1);
  // S7: pool -> bufB (F=128)
  gcn_pool<<<dim3(NP, GB), 256, 0, stream>>>(bufA, adjP, bufB, 128);
  // S8: bufA = bufB @ gc3_wp                    K=128 F=96
  gcn_wmma_gemm_f32<<<dim3(MT, 3, 1), 32, 0, stream>>>(bufB, LDH, 0, gc3_wp, 96, 0,
                                                       bufA, LDH, 0, nullptr, 32, 0);
  // S9: bufB = relu(adjP @ bufA + gc3_b)        F=96
  gcn_wmma_gemm_f32<<<dim3(AT, 3, GB), 32, 0, stream>>>(adjP, NP, SA, bufA, LDH, SH,
                                                        bufB, LDH, SH, gc3_bp, 40, 1);
  // S10: pool -> bufA (F=96; only cols <75 feed the head)
  gcn_pool<<<dim3(NP, GB), 256, 0, stream>>>(bufB, adjP, bufA, 96);
  // S11: head -> out
  gcn_head<<<dim3(GB), 256, 0, stream>>>(bufA, adjP, gc4_w, gc4_b,
                                         fc1_w, fc1_b, fin_w, fin_b, out);
}